// Offset2D_17772574671403
// MI455X (gfx1250) — compile-verified
//
#include <hip/hip_runtime.h>

// MI455X / gfx1250, wave32. Fused 1x1-conv (via V_WMMA_F32_16X16X4_F32) +
// offset/attention epilogue + scatter-add, then a finalize divide kernel.
// Streaming outputs (destination/offset/out) use non-temporal hints so they
// don't evict the x tiles and the L2-resident atomic accumulators.

typedef __attribute__((ext_vector_type(2))) float v2f;
typedef __attribute__((ext_vector_type(4))) float v4f;
typedef __attribute__((ext_vector_type(8))) float v8f;

#define B_   32
#define C_   256
#define H_   128
#define W_   128
#define HW_  (H_ * W_)        // 16384 = 2^14
#define DH_  64
#define DW_  64
#define DHW_ (DH_ * DW_)      // 4096 = 2^12
#define EPS_ 1e-5f

// One wave handles one tile of 16 consecutive pixels (never crosses a row:
// W=128 is a multiple of 16). 8 waves per 256-thread block.
__global__ __launch_bounds__(256) void conv_scatter_kernel(
    const float* __restrict__ x,        // (B,C,H,W)
    const float* __restrict__ conv_w,   // (3,C)
    const float* __restrict__ conv_b,   // (3,)
    float*       __restrict__ out_feat, // (B,C,DH,DW) pre-zeroed, atomic accum
    float*       __restrict__ out_off,  // (B,2,H,W)
    int*         __restrict__ out_dest, // (B,C,H,W) int32
    float*       __restrict__ att_sum)  // (B,DH*DW) pre-zeroed, atomic accum
{
    // Padded weights: ldsB[c][n] = (n<3) ? conv_w[n][c] : 0   (16 KB)
    __shared__ float ldsB[C_ * 16];
    // Per-wave conv outputs: 16 pixels x {oy, ox, oz} (padded to 4)
    __shared__ float ldsOA[8][16][4];

    const int tid = threadIdx.x;
    for (int i = tid; i < C_ * 16; i += 256) {
        const int c = i >> 4, n = i & 15;
        ldsB[i] = (n < 3) ? conv_w[n * C_ + c] : 0.0f;
    }
    __syncthreads();

    const int wave = tid >> 5;
    const int lane = tid & 31;
    const int m    = lane & 15;   // pixel-in-tile (A rows) == B column n
    const int kh   = lane >> 4;   // K-half selector

    const int tile = blockIdx.x * 8 + wave;
    const int p0   = tile * 16;                // linear pixel index (b,h,w)
    const int b    = p0 >> 14;                 // / HW
    const int hw   = p0 & (HW_ - 1);
    const int h    = hw >> 7;                  // / W
    const int w0   = hw & (W_ - 1);

    // Base pointer for this lane's pixel, channel stride = HW
    const float* __restrict__ px = x + (size_t)b * C_ * HW_ + h * W_ + w0 + m;

    // -------- 1x1 conv as GEMM: (16 x 256) * (256 x 16[3 used]) ----------
    v8f acc = {};
    #pragma unroll 4
    for (int k0 = 0; k0 < C_; k0 += 4) {
        const int c0 = k0 + 2 * kh;
        v2f a, bm;
        a.x  = px[(size_t)c0 * HW_];           // A[m][2kh]
        a.y  = px[(size_t)(c0 + 1) * HW_];     // A[m][2kh+1]
        bm.x = ldsB[c0 * 16 + m];              // B[2kh][n=m]
        bm.y = ldsB[(c0 + 1) * 16 + m];        // B[2kh+1][n=m]
        acc = __builtin_amdgcn_wmma_f32_16x16x4_f32(
            false, a, false, bm, (short)0, acc, false, false);
    }

    // -------- redistribute columns 0..2 to per-pixel across lanes --------
    if (m < 3) {
        #pragma unroll
        for (int v = 0; v < 8; ++v)
            ldsOA[wave][v + 8 * kh][m] = acc[v];   // row M = v + 8*kh
    }
    __syncthreads();

    const float oy  = ldsOA[wave][m][0] + conv_b[0];
    const float ox  = ldsOA[wave][m][1] + conv_b[1];
    const float att = expf(ldsOA[wave][m][2] + conv_b[2]);

    const float gy = (float)h * (1.0f / H_);
    const float gx = (float)(w0 + m) * (1.0f / W_);
    const float cy = fminf(fmaxf(gy + oy, 0.0f), 1.0f - EPS_);
    const float cx = fminf(fmaxf(gx + ox, 0.0f), 1.0f - EPS_);
    const int   dy = (int)floorf(cy * (float)DH_);
    const int   dx = (int)floorf(cx * (float)DW_);
    const int cell = dy * DW_ + dx;

    if (lane < 16) {
        const size_t obase = (size_t)b * 2 * HW_ + h * W_ + w0 + m;
        __builtin_nontemporal_store(oy, &out_off[obase]);        // write-once
        __builtin_nontemporal_store(ox, &out_off[obase + HW_]);  // write-once
        atomicAdd(&att_sum[b * DHW_ + cell], att);
    }

    // -------- scatter: feat[b,c,cell] += x*att ; destination[b,c,p] ------
    const int     dbase = b * C_ * DHW_ + cell;           // fits int32
    float* __restrict__ fptr = out_feat + (size_t)b * C_ * DHW_ + cell;
    int*   __restrict__ dptr = out_dest + (size_t)b * C_ * HW_ + h * W_ + w0 + m;

    #pragma unroll 2
    for (int ci = 0; ci < C_; ci += 2) {
        const int c = ci + kh;                 // kh=0 -> even, kh=1 -> odd
        const float val = px[(size_t)c * HW_]; // WGP$/L2 hit (just read above)
        atomicAdd(&fptr[(size_t)c * DHW_], val * att);
        // 537 MB streaming output: NT so it doesn't evict x / out_feat lines
        __builtin_nontemporal_store(dbase + c * DHW_, &dptr[(size_t)c * HW_]);
    }
}

__global__ __launch_bounds__(256) void finalize_kernel(
    float* __restrict__ out, const float* __restrict__ att_sum)
{
    const size_t i4 = ((size_t)blockIdx.x * 256 + threadIdx.x) * 4;
    const int b   = (int)(i4 >> 20);           // / (C*DHW) = / 2^20
    const int pos = (int)(i4 & (DHW_ - 1));    // 4-aligned, same b for all 4

    v4f f = __builtin_nontemporal_load((const v4f*)&out[i4]);
    const float* as = &att_sum[b * DHW_ + pos];
    v4f r;
    r.x = f.x / (as[0] + EPS_);
    r.y = f.y / (as[1] + EPS_);
    r.z = f.z / (as[2] + EPS_);
    r.w = f.w / (as[3] + EPS_);
    __builtin_nontemporal_store(r, (v4f*)&out[i4]);
}

extern "C" void kernel_launch(void* const* d_in, const int* in_sizes, int n_in,
                              void* d_out, int out_size, void* d_ws, size_t ws_size,
                              hipStream_t stream)
{
    const float* x      = (const float*)d_in[0];
    const float* conv_w = (const float*)d_in[1];
    const float* conv_b = (const float*)d_in[2];

    // d_out = [out (B*C*DHW) | offset (B*2*HW) | destination (B*C*HW) int32]
    float* out      = (float*)d_out;
    float* out_off  = out + (size_t)B_ * C_ * DHW_;
    int*   out_dest = (int*)(out_off + (size_t)B_ * 2 * HW_);
    float* att_sum  = (float*)d_ws;            // B*DHW floats

    hipMemsetAsync(out,     0, (size_t)B_ * C_ * DHW_ * sizeof(float), stream);
    hipMemsetAsync(att_sum, 0, (size_t)B_ * DHW_ * sizeof(float),      stream);

    const int n_tiles = B_ * HW_ / 16;         // 32768 waves
    conv_scatter_kernel<<<n_tiles / 8, 256, 0, stream>>>(
        x, conv_w, conv_b, out, out_off, out_dest, att_sum);

    const size_t n_out = (size_t)B_ * C_ * DHW_;
    finalize_kernel<<<(unsigned)(n_out / 1024), 256, 0, stream>>>(out, att_sum);
}